// VolumeRenderer_75247827026376
// MI455X (gfx1250) — compile-verified
//
#include <hip/hip_runtime.h>
#include <hip/hip_bf16.h>
#include <stdint.h>

// Volume rendering (NeRF ray marching), MI455X / gfx1250.
// One ray per lane (wave32). Per 64-ray block, all input data is contiguous
// in global memory, so we DMA it into LDS with coalesced
// global_load_async_to_lds_b128 (ASYNCcnt-tracked CDNA5 async path), with
// padded LDS strides to kill bank conflicts, then run the serial
// transmittance scan out of LDS.

#define NSAMP   48
#define RAYS_PB 64          // rays (threads) per block
#define ZPAD    52          // padded per-ray stride for z / sigma (floats), 16B-mult
#define RGBPAD  148         // padded per-ray stride for rgb (floats), 16B-mult

__device__ __forceinline__ unsigned lds_addr_of(const void* p) {
    // Generic pointer into the shared aperture: low 32 bits are the LDS byte
    // address used by DS / async-to-LDS instructions.
    return (unsigned)(unsigned long long)p;
}

__device__ __forceinline__ void async_load_b128(unsigned lds_byte, const float* gaddr) {
    asm volatile("global_load_async_to_lds_b128 %0, %1, off"
                 :: "v"(lds_byte), "v"(gaddr) : "memory");
}

__global__ void __launch_bounds__(RAYS_PB)
volrender_kernel(const float* __restrict__ rgb,
                 const float* __restrict__ sigma,
                 const float* __restrict__ zv,
                 float* __restrict__ out_rgb,
                 float* __restrict__ out_depth)
{
    __shared__ __align__(16) float s_z  [RAYS_PB * ZPAD];    // 13312 B
    __shared__ __align__(16) float s_sig[RAYS_PB * ZPAD];    // 13312 B
    __shared__ __align__(16) float s_rgb[RAYS_PB * RGBPAD];  // 37888 B  -> 63 KB total

    const int t = threadIdx.x;
    const long long ray0 = (long long)blockIdx.x * RAYS_PB;

    const float* gz = zv    + ray0 * NSAMP;       // 64*48   floats, contiguous
    const float* gs = sigma + ray0 * NSAMP;       // 64*48   floats, contiguous
    const float* gr = rgb   + ray0 * NSAMP * 3;   // 64*144  floats, contiguous

    // ---- async DMA: global (flat, coalesced 16B/lane) -> LDS (padded) ----
    // z: 12 b128 units per ray. Flat unit u -> global byte 16*u exactly.
    #pragma unroll
    for (int i = 0; i < 12; ++i) {
        int u = i * RAYS_PB + t;
        int ray = u / 12, part = u % 12;
        async_load_b128(lds_addr_of(&s_z[ray * ZPAD + part * 4]), gz + u * 4);
    }
    // sigma: same shape as z.
    #pragma unroll
    for (int i = 0; i < 12; ++i) {
        int u = i * RAYS_PB + t;
        int ray = u / 12, part = u % 12;
        async_load_b128(lds_addr_of(&s_sig[ray * ZPAD + part * 4]), gs + u * 4);
    }
    // rgb: 36 b128 units per ray.
    #pragma unroll
    for (int i = 0; i < 36; ++i) {
        int u = i * RAYS_PB + t;
        int ray = u / 36, part = u % 36;
        async_load_b128(lds_addr_of(&s_rgb[ray * RGBPAD + part * 4]), gr + u * 4);
    }

    asm volatile("s_wait_asynccnt 0" ::: "memory");
    __syncthreads();

    // ---- per-ray serial scan (exclusive cumprod transmittance) ----
    const float* zr = &s_z  [t * ZPAD];
    const float* sr = &s_sig[t * ZPAD];
    const float* rr = &s_rgb[t * RGBPAD];

    float T = 1.0f;
    float accr = 0.0f, accg = 0.0f, accb = 0.0f, accd = 0.0f;
    float zc = zr[0];

    #pragma unroll
    for (int s = 0; s < NSAMP; ++s) {
        float zn   = (s < NSAMP - 1) ? zr[s + 1] : 0.0f;
        float dist = (s < NSAMP - 1) ? (zn - zc) : 1e10f;
        float alpha = 1.0f - __expf(-sr[s] * dist);
        float w = alpha * T;
        T *= (1.0f - alpha + 1e-10f);
        accr += w * rr[s * 3 + 0];
        accg += w * rr[s * 3 + 1];
        accb += w * rr[s * 3 + 2];
        accd += w * zc;
        zc = zn;
    }

    const long long ray = ray0 + t;
    out_rgb[ray * 3 + 0] = accr;
    out_rgb[ray * 3 + 1] = accg;
    out_rgb[ray * 3 + 2] = accb;
    out_depth[ray]       = accd;
}

extern "C" void kernel_launch(void* const* d_in, const int* in_sizes, int n_in,
                              void* d_out, int out_size, void* d_ws, size_t ws_size,
                              hipStream_t stream) {
    const float* rgb   = (const float*)d_in[0];  // (B, HW, N, 3)
    const float* sigma = (const float*)d_in[1];  // (B, HW, N, 1)
    const float* zv    = (const float*)d_in[2];  // (B, HW, N)

    const long long nrays = (long long)in_sizes[2] / NSAMP;  // B*HW = 262144
    float* out_rgb   = (float*)d_out;
    float* out_depth = out_rgb + nrays * 3;

    const int blocks = (int)(nrays / RAYS_PB);   // 4096, exact
    volrender_kernel<<<blocks, RAYS_PB, 0, stream>>>(rgb, sigma, zv, out_rgb, out_depth);
}